// ResNetFeatureExtractor_proposedStructuredPooling_69028714381988
// MI455X (gfx1250) — compile-verified
//
#include <hip/hip_runtime.h>

typedef __attribute__((ext_vector_type(2))) float v2f;
typedef __attribute__((ext_vector_type(8))) float v8f;

#define B_ 32
#define M_ 32
#define C_ 1024
#define R_ 28

// Workspace layout (floats):
//   rowT : [B][R][C]  = B matrix per batch (K x N), offset 0        (3,670,016 B)
//   colS : [B][M][R]  = A matrix per batch (M x K), after rowT      (  114,688 B)

// ---------------------------------------------------------------------------
// col[b,m,k] = sum_i Masks[b,m,i,k]   (coalesced: k fastest across threads)
// ---------------------------------------------------------------------------
__global__ __launch_bounds__(256) void mask_colsum_kernel(
    const float* __restrict__ Masks, float* __restrict__ colS) {
  int t = blockIdx.x * 256 + threadIdx.x;      // flat (b,m,k), total 28672
  int k  = t % R_;
  int bm = t / R_;                             // b*M_ + m
  const float* p = Masks + (size_t)(bm * R_) * R_ + k;
  float s = 0.f;
#pragma unroll
  for (int i = 0; i < R_; ++i) s += p[i * R_];
  colS[t] = s;                                 // layout (b,m,k) == flat t
}

// ---------------------------------------------------------------------------
// rowT[b,i,c] = sum_j F[b,c,i,j]   (each thread streams one contiguous row)
// ---------------------------------------------------------------------------
__global__ __launch_bounds__(256) void feat_rowsum_kernel(
    const float* __restrict__ F, float* __restrict__ rowT) {
  int t = blockIdx.x * 256 + threadIdx.x;      // flat (b,c,i), total 917504
  const float4* p = (const float4*)(F + (size_t)t * R_);  // 112B rows, 16B aligned
  float s = 0.f;
#pragma unroll
  for (int q = 0; q < 7; ++q) {
    float4 v = p[q];
    s += (v.x + v.y) + (v.z + v.w);
  }
  int i  = t % R_;
  int bc = t / R_;
  int c  = bc % C_;
  int b  = bc / C_;
  rowT[((size_t)b * R_ + i) * C_ + c] = s;     // transposed store -> B matrix layout
}

// ---------------------------------------------------------------------------
// S[b,m,c] = (1/784) * sum_k colS[b,m,k] * rowT[b,k,c]
// One wave per 16x16 tile, 7x V_WMMA_F32_16X16X4_F32 (K = 28).
// ---------------------------------------------------------------------------
__global__ __launch_bounds__(256) void wmma_gemm_kernel(
    const float* __restrict__ colS, const float* __restrict__ rowT,
    float* __restrict__ S) {
  int wave = (blockIdx.x * 256 + threadIdx.x) >> 5;   // tile id 0..4095
  int lane = threadIdx.x & 31;
  int nt = wave & 63;          // 64 tiles along C
  int mt = (wave >> 6) & 1;    // 2 tiles along M
  int b  = wave >> 7;          // batch

  int half = lane >> 4;        // 0: lanes 0-15, 1: lanes 16-31
  int l16  = lane & 15;
  int m = mt * 16 + l16;       // A row for this lane
  int n = nt * 16 + l16;       // B column for this lane

  const float* ca = colS + (size_t)(b * M_ + m) * R_;       // A row (len 28)
  const float* rb = rowT + (size_t)b * R_ * C_ + n;         // B col (stride C_)

  v8f acc = {0.f, 0.f, 0.f, 0.f, 0.f, 0.f, 0.f, 0.f};
#pragma unroll
  for (int k0 = 0; k0 < R_; k0 += 4) {
    int kk = k0 + 2 * half;                    // ISA A/B lane layout for 16x4 f32
    v2f a;  a.x  = ca[kk];        a.y  = ca[kk + 1];
    v2f bb; bb.x = rb[kk * C_];   bb.y = rb[(kk + 1) * C_];
    acc = __builtin_amdgcn_wmma_f32_16x16x4_f32(
        /*neg_a=*/false, a, /*neg_b=*/false, bb,
        /*c_mod=*/(short)0, acc, /*reuse_a=*/false, /*reuse_b=*/false);
  }

  const float inv = 1.0f / (float)(R_ * R_);
#pragma unroll
  for (int v = 0; v < 8; ++v) {
    int row = mt * 16 + v + 8 * half;          // C/D layout: VGPR v -> M=v / v+8
    S[((size_t)b * M_ + row) * C_ + n] = acc[v] * inv;
  }
}

// ---------------------------------------------------------------------------
extern "C" void kernel_launch(void* const* d_in, const int* in_sizes, int n_in,
                              void* d_out, int out_size, void* d_ws, size_t ws_size,
                              hipStream_t stream) {
  const float* F     = (const float*)d_in[0];   // (B, C, R, R)
  const float* Masks = (const float*)d_in[1];   // (B, M, R, R)
  float* S = (float*)d_out;                     // (B, M, C)

  float* rowT = (float*)d_ws;                              // B*R*C floats
  float* colS = rowT + (size_t)B_ * R_ * C_;               // B*M*R floats

  // Independent reductions; stream order serializes them before the GEMM.
  mask_colsum_kernel<<<(B_ * M_ * R_) / 256, 256, 0, stream>>>(Masks, colS);
  feat_rowsum_kernel<<<(B_ * C_ * R_) / 256, 256, 0, stream>>>(F, rowT);

  // 4096 tiles (32 batches x 2 mTiles x 64 nTiles), 8 waves per 256-thread block.
  wmma_gemm_kernel<<<(B_ * 2 * 64) / 8, 256, 0, stream>>>(colS, rowT, S);
}